// BayesianUncertaintyDistance_23141283791061
// MI455X (gfx1250) — compile-verified
//
#include <hip/hip_runtime.h>
#include <hip/hip_bf16.h>

// ---------------------------------------------------------------------------
// BayesianUncertaintyDistance for MI455X (gfx1250, wave32, WMMA).
// D=256, Q=8192, P=2048, S=10.  ~97 GFLOP f32 matmul dominates; outputs 134MB.
// All GEMM work uses V_WMMA_F32_16X16X4_F32 (f32-exact matrix path).
// dist[S,Q,P] is never materialized: distance + mean/std accumulation is fused
// into the cross-GEMM epilogue, in registers.
// Main kernel is 2x2 register-blocked (32x32 tile per wave): 4 WMMAs per
// 4 b64 loads -> 1:1 vmem:wmma issue ratio, ~8 FLOP/B from cache.
// ---------------------------------------------------------------------------

typedef __attribute__((ext_vector_type(2))) float v2f;
typedef __attribute__((ext_vector_type(8))) float v8f;

#define DDIM 256
#define QDIM 8192
#define PDIM 2048

__device__ __forceinline__ float softplus_f(float x) {
  // log(1+exp(x)), overflow-safe
  return (x > 20.0f) ? x : log1pf(__expf(x));
}

// W[s,d,e] = mu[d,e] + eps_w[s,d,e] * softplus(rho[d,e])
__global__ void prep_weights_kernel(const float* __restrict__ mu,
                                    const float* __restrict__ rho,
                                    const float* __restrict__ eps_w,
                                    float* __restrict__ W, int total) {
  int i = blockIdx.x * blockDim.x + threadIdx.x;
  if (i >= total) return;
  int j = i % (DDIM * DDIM);
  W[i] = mu[j] + eps_w[i] * softplus_f(rho[j]);
}

// bias[s,e] = bmu[e] + eps_b[s,e] * softplus(brho[e])
__global__ void prep_bias_kernel(const float* __restrict__ bmu,
                                 const float* __restrict__ brho,
                                 const float* __restrict__ eps_b,
                                 float* __restrict__ bias, int total) {
  int i = blockIdx.x * blockDim.x + threadIdx.x;
  if (i >= total) return;
  int j = i % DDIM;
  bias[i] = bmu[j] + eps_b[i] * softplus_f(brho[j]);
}

// pn[p] = sum_e proto[p,e]^2
__global__ void prep_pn_kernel(const float* __restrict__ proto,
                               float* __restrict__ pn) {
  int p = blockIdx.x * blockDim.x + threadIdx.x;
  if (p >= PDIM) return;
  const float4* row = (const float4*)(proto + (size_t)p * DDIM);
  float s = 0.f;
#pragma unroll 4
  for (int i = 0; i < DDIM / 4; ++i) {
    float4 v = row[i];
    s += v.x * v.x + v.y * v.y + v.z * v.z + v.w * v.w;
  }
  pn[p] = s;
}

// qt[s,q,e] = sum_d query[q,d] * W[s,d,e] + bias[s,e]
// One wave per 16x16 output tile; K=256 via 64x wmma_f32_16x16x4_f32.
// Grid is exact (no early exit) so EXEC is all-ones for every WMMA.
__global__ void qt_gemm_kernel(const float* __restrict__ query,
                               const float* __restrict__ W,
                               const float* __restrict__ bias,
                               float* __restrict__ qt, int S) {
  const int wave = threadIdx.x >> 5;
  const int lane = threadIdx.x & 31;
  const int half = lane >> 4;   // 0: K in {0,1}, 1: K in {2,3}
  const int l16  = lane & 15;

  const int QT = QDIM / 16, ET = DDIM / 16;
  int t    = blockIdx.x * 8 + wave;
  int s    = t / (QT * ET);
  int rem  = t % (QT * ET);
  int qbase = (rem / ET) * 16;
  int ebase = (rem % ET) * 16;

  // A frag: query[qbase + l16][d0 + 2*half + v]  (contiguous pair -> b64 load)
  const float* arow  = query + (size_t)(qbase + l16) * DDIM + 2 * half;
  // B frag: W[s][d0 + 2*half + v][ebase + l16]   (row-strided pair)
  const float* Wbase = W + (size_t)s * DDIM * DDIM + ebase + l16;

  v8f acc = {0.f, 0.f, 0.f, 0.f, 0.f, 0.f, 0.f, 0.f};
#pragma unroll 8
  for (int d0 = 0; d0 < DDIM; d0 += 4) {
    v2f a = *(const v2f*)(arow + d0);
    v2f b;
    const float* bp = Wbase + (size_t)(d0 + 2 * half) * DDIM;
    b.x = bp[0];
    b.y = bp[DDIM];
    acc = __builtin_amdgcn_wmma_f32_16x16x4_f32(false, a, false, b,
                                                (short)0, acc, false, false);
  }

  float bb = bias[s * DDIM + ebase + l16];
  // C/D layout: VGPR v, lane l -> row (qbase + v + 8*half), col (ebase + l16)
  float* orow =
      qt + ((size_t)s * QDIM + qbase + 8 * half) * DDIM + ebase + l16;
#pragma unroll
  for (int v = 0; v < 8; ++v) orow[(size_t)v * DDIM] = acc[v] + bb;
}

// qn[s*Q + q] = sum_e qt[s,q,e]^2
__global__ void qn_kernel(const float* __restrict__ qt,
                          float* __restrict__ qn, int total) {
  int i = blockIdx.x * blockDim.x + threadIdx.x;
  if (i >= total) return;
  const float4* row = (const float4*)(qt + (size_t)i * DDIM);
  float s = 0.f;
#pragma unroll 4
  for (int k = 0; k < DDIM / 4; ++k) {
    float4 v = row[k];
    s += v.x * v.x + v.y * v.y + v.z * v.z + v.w * v.w;
  }
  qn[i] = s;
}

// Per 32x32 (q,p) tile (2x2 blocking of 16x16 WMMA tiles), one wave per tile.
// For each s: cross = qt[s] @ proto^T via WMMA over K=256 (4 wmma / 4 loads),
// then dist = sqrt(max(qn+pn-2*cross,0)); accumulate sum/sumsq over s;
// emit mean and unbiased std.
__global__ void dist_main_kernel(const float* __restrict__ qt,
                                 const float* __restrict__ qn,
                                 const float* __restrict__ pn,
                                 const float* __restrict__ proto,
                                 float* __restrict__ out, int S) {
  const int wave = threadIdx.x >> 5;
  const int lane = threadIdx.x & 31;
  const int half = lane >> 4;
  const int l16  = lane & 15;

  const int PT = PDIM / 32;                // 64 p-tiles of 32
  int t     = blockIdx.x * 8 + wave;       // 16384 tiles total
  int qbase = (t / PT) * 32;
  int pbase = (t % PT) * 32;

  // B frags (proto^T): B[K=e][N=p] = proto[p][e]; lane holds contiguous e-pair.
  const float* brow0 = proto + (size_t)(pbase + l16) * DDIM + 2 * half;
  const float* brow1 = brow0 + (size_t)16 * DDIM;
  float pn0 = pn[pbase + l16];
  float pn1 = pn[pbase + 16 + l16];

  float sum[2][2][8], sq[2][2][8];         // [q-subtile][p-subtile][vgpr]
#pragma unroll
  for (int i = 0; i < 2; ++i)
#pragma unroll
    for (int j = 0; j < 2; ++j)
#pragma unroll
      for (int v = 0; v < 8; ++v) { sum[i][j][v] = 0.f; sq[i][j][v] = 0.f; }

  for (int s = 0; s < S; ++s) {
    const float* arow0 =
        qt + ((size_t)s * QDIM + qbase + l16) * DDIM + 2 * half;
    const float* arow1 = arow0 + (size_t)16 * DDIM;

    v8f acc00 = {0.f,0.f,0.f,0.f,0.f,0.f,0.f,0.f};
    v8f acc01 = acc00, acc10 = acc00, acc11 = acc00;
#pragma unroll 4
    for (int e0 = 0; e0 < DDIM; e0 += 4) {
      v2f a0 = *(const v2f*)(arow0 + e0);
      v2f a1 = *(const v2f*)(arow1 + e0);
      v2f b0 = *(const v2f*)(brow0 + e0);
      v2f b1 = *(const v2f*)(brow1 + e0);
      acc00 = __builtin_amdgcn_wmma_f32_16x16x4_f32(false, a0, false, b0,
                                                    (short)0, acc00, false, false);
      acc01 = __builtin_amdgcn_wmma_f32_16x16x4_f32(false, a0, false, b1,
                                                    (short)0, acc01, false, false);
      acc10 = __builtin_amdgcn_wmma_f32_16x16x4_f32(false, a1, false, b0,
                                                    (short)0, acc10, false, false);
      acc11 = __builtin_amdgcn_wmma_f32_16x16x4_f32(false, a1, false, b1,
                                                    (short)0, acc11, false, false);
    }

    // qn for this tile's rows: contiguous 8-float groups, 32B-aligned.
    const float4* qn0 =
        (const float4*)(qn + (size_t)s * QDIM + qbase + 8 * half);
    const float4* qn1 =
        (const float4*)(qn + (size_t)s * QDIM + qbase + 16 + 8 * half);
    float4 a = qn0[0], b = qn0[1], c = qn1[0], d = qn1[1];
    float qs0[8] = {a.x, a.y, a.z, a.w, b.x, b.y, b.z, b.w};
    float qs1[8] = {c.x, c.y, c.z, c.w, d.x, d.y, d.z, d.w};

#pragma unroll
    for (int v = 0; v < 8; ++v) {
      float d00 = sqrtf(fmaxf(qs0[v] + pn0 - 2.0f * acc00[v], 0.f));
      float d01 = sqrtf(fmaxf(qs0[v] + pn1 - 2.0f * acc01[v], 0.f));
      float d10 = sqrtf(fmaxf(qs1[v] + pn0 - 2.0f * acc10[v], 0.f));
      float d11 = sqrtf(fmaxf(qs1[v] + pn1 - 2.0f * acc11[v], 0.f));
      sum[0][0][v] += d00; sq[0][0][v] += d00 * d00;
      sum[0][1][v] += d01; sq[0][1][v] += d01 * d01;
      sum[1][0][v] += d10; sq[1][0][v] += d10 * d10;
      sum[1][1][v] += d11; sq[1][1][v] += d11 * d11;
    }
  }

  float invS  = 1.0f / (float)S;
  float invS1 = 1.0f / (float)(S - 1);
#pragma unroll
  for (int i = 0; i < 2; ++i) {
#pragma unroll
    for (int j = 0; j < 2; ++j) {
      size_t rowoff = (size_t)(qbase + 16 * i + 8 * half) * PDIM +
                      (pbase + 16 * j + l16);
      float* mrow = out + rowoff;
      float* srow = out + (size_t)QDIM * PDIM + rowoff;
#pragma unroll
      for (int v = 0; v < 8; ++v) {
        float mean = sum[i][j][v] * invS;
        float var  = (sq[i][j][v] - (float)S * mean * mean) * invS1;
        mrow[(size_t)v * PDIM] = mean;
        srow[(size_t)v * PDIM] = sqrtf(fmaxf(var, 0.f));
      }
    }
  }
}

extern "C" void kernel_launch(void* const* d_in, const int* in_sizes, int n_in,
                              void* d_out, int out_size, void* d_ws,
                              size_t ws_size, hipStream_t stream) {
  const float* query = (const float*)d_in[0];  // [Q, D]
  const float* proto = (const float*)d_in[1];  // [P, D]
  const float* wmu   = (const float*)d_in[2];  // [D, D]
  const float* wrho  = (const float*)d_in[3];  // [D, D]
  const float* bmu   = (const float*)d_in[4];  // [D]
  const float* brho  = (const float*)d_in[5];  // [D]
  const float* eps_w = (const float*)d_in[6];  // [S, D, D]
  const float* eps_b = (const float*)d_in[7];  // [S, D]
  int S = in_sizes[6] / (DDIM * DDIM);         // = 10

  // Workspace layout (f32), ~87 MB total:
  float* ws   = (float*)d_ws;
  float* W    = ws;                                    // S*D*D
  float* bias = W + (size_t)S * DDIM * DDIM;           // S*D
  float* pn   = bias + (size_t)S * DDIM;               // P
  float* qt   = pn + PDIM;                             // S*Q*D
  float* qn   = qt + (size_t)S * QDIM * DDIM;          // S*Q

  prep_weights_kernel<<<(S * DDIM * DDIM + 255) / 256, 256, 0, stream>>>(
      wmu, wrho, eps_w, W, S * DDIM * DDIM);
  prep_bias_kernel<<<(S * DDIM + 255) / 256, 256, 0, stream>>>(
      bmu, brho, eps_b, bias, S * DDIM);
  prep_pn_kernel<<<(PDIM + 255) / 256, 256, 0, stream>>>(proto, pn);

  int qt_tiles = S * (QDIM / 16) * (DDIM / 16);  // 81920 -> 10240 blocks
  qt_gemm_kernel<<<qt_tiles / 8, 256, 0, stream>>>(query, W, bias, qt, S);

  qn_kernel<<<(S * QDIM + 255) / 256, 256, 0, stream>>>(qt, qn, S * QDIM);

  int tiles = (QDIM / 32) * (PDIM / 32);  // 16384 -> 2048 blocks
  dist_main_kernel<<<tiles / 8, 256, 0, stream>>>(qt, qn, pn, proto,
                                                  (float*)d_out, S);
}